// LiquidRingMoELayer_51531017617704
// MI455X (gfx1250) — compile-verified
//
#include <hip/hip_runtime.h>
#include <math.h>

typedef float v2f __attribute__((ext_vector_type(2)));
typedef float v8f __attribute__((ext_vector_type(8)));

#define WMMA_F32(a, b, c) \
  __builtin_amdgcn_wmma_f32_16x16x4_f32(false, (a), false, (b), (short)0, (c), false, false)

__device__ __forceinline__ float gelu_exact(float x) {
  return 0.5f * x * (1.0f + erff(x * 0.70710678118654752440f));
}

// Truncate generic pointer to LDS byte offset (low 32 bits of the shared
// aperture address are the wave-relative LDS address per CDNA5 aperture rules).
__device__ __forceinline__ unsigned lds_off(const void* p) {
  return (unsigned)(size_t)p;
}

// ---------------------------------------------------------------------------
// fp32 WMMA GEMM: C[M,N] = epi(A[M,K] @ B[K,N] + bias)
// Block = 256 threads (8 waves) computes a 128x64 tile; wave w owns rows
// [16w,16w+16). The B strip (16x64 fp32 chunk, 4 KB) is shared by all waves:
// staged into LDS by the gfx1250 async copy engine (one
// global_load_async_to_lds_b128 per wave per chunk), double-buffered with
// s_wait_asynccnt so DMA of chunk c+1 overlaps WMMA of chunk c.
// EPI: 0 = none, 1 = 0.1*tanh, 2 = exact GELU
// ---------------------------------------------------------------------------
template <int EPI>
__global__ void gemm128x64(const float* __restrict__ A, const float* __restrict__ B,
                           const float* __restrict__ bias, float* __restrict__ C,
                           int M, int N, int K) {
  __shared__ __align__(16) float sB[2][16 * 64];

  const int lane = threadIdx.x & 31;
  const int wave = threadIdx.x >> 5;
  const int nStrips = N >> 6;
  const int mBlk = blockIdx.x / nStrips;
  const int nStrip = blockIdx.x - mBlk * nStrips;
  const int m0 = (mBlk << 7) + (wave << 4);
  const int n0 = nStrip << 6;
  const int lm = lane & 15;            // row (A) / col (B,D) within tile
  const int kh = (lane >> 4) << 1;     // k-half: 0 or 2

  // Async-copy lane assignment: wave w moves rows 2w,2w+1 of each 16x64 chunk.
  // lane 0-15 -> row 2w (16B each), lane 16-31 -> row 2w+1.
  const int brow = (wave << 1) + (lane >> 4);
  const int bcol = (lane & 15) << 2;   // 4 floats / lane
  unsigned voff = (unsigned)(((size_t)brow * N + n0 + bcol) * 4);
  const unsigned vstep = (unsigned)(16u * (unsigned)N * 4u);
  const unsigned ldsDst0 = lds_off(&sB[0][brow * 64 + bcol]);
  const unsigned ldsDst1 = lds_off(&sB[1][brow * 64 + bcol]);

  // Preload chunk 0 into buffer 0.
  asm volatile("global_load_async_to_lds_b128 %0, %1, %2"
               :: "v"(ldsDst0), "v"(voff), "s"(B) : "memory");
  voff += vstep;

  v8f acc[4] = {};
  const float* Ar = A + (size_t)(m0 + lm) * K + kh;
  const int nChunks = K >> 4;
  for (int c = 0; c < nChunks; ++c) {
    if (c + 1 < nChunks) {
      // Prefetch chunk c+1 into the other buffer (its previous readers all
      // passed the barrier at the end of iteration c-1).
      asm volatile("global_load_async_to_lds_b128 %0, %1, %2"
                   :: "v"((c & 1) ? ldsDst0 : ldsDst1), "v"(voff), "s"(B) : "memory");
      voff += vstep;
      asm volatile("s_wait_asynccnt 1" ::: "memory");  // chunk c landed
    } else {
      asm volatile("s_wait_asynccnt 0" ::: "memory");
    }
    __syncthreads();                                    // all waves' DMA done

    const float* bb = sB[c & 1];
    const int kbase = c << 4;
#pragma unroll
    for (int ks = 0; ks < 16; ks += 4) {
      v2f a;
      a.x = Ar[kbase + ks];
      a.y = Ar[kbase + ks + 1];
      const float* Bp = bb + (ks + kh) * 64 + lm;
#pragma unroll
      for (int s = 0; s < 4; ++s) {
        v2f bv;
        bv.x = Bp[s * 16];
        bv.y = Bp[64 + s * 16];
        acc[s] = WMMA_F32(a, bv, acc[s]);
      }
    }
    __syncthreads();  // protect buffer (c&1) before it is overwritten
  }

  const int mo = (lane >> 4) << 3;     // D-matrix row offset: 0 or 8
#pragma unroll
  for (int s = 0; s < 4; ++s) {
    const int n = n0 + (s << 4) + lm;
    const float bv = bias ? bias[n] : 0.0f;
#pragma unroll
    for (int r = 0; r < 8; ++r) {
      float v = acc[s][r] + bv;
      if (EPI == 1) v = 0.1f * tanhf(v);
      if (EPI == 2) v = gelu_exact(v);
      C[(size_t)(m0 + mo + r) * N + n] = v;
    }
  }
}

// ---------------------------------------------------------------------------
// Attention for one (batch, head, 16-query tile). head_dim = 16.
// Phase 1: scores 16x256 via WMMA (Q tiles x K^T), scaled + key-masked -> LDS
// Phase 2: row softmax (wave-per-2-rows, shfl reductions)
// Phase 3: P @ V via WMMA, K-split across 8 waves, LDS reduction
// ---------------------------------------------------------------------------
__global__ void attn16(const float* __restrict__ Q, const float* __restrict__ Km,
                       const float* __restrict__ V, const float* __restrict__ kept,
                       float* __restrict__ AO, int S, int D, int H) {
  const int QT = S >> 4;
  int idx = blockIdx.x;
  const int qt = idx % QT; idx /= QT;
  const int h  = idx % H;  idx /= H;
  const int b  = idx;
  const int hoff = h << 4;
  const int base = b * S;
  const int q0 = qt << 4;

  const int lane = threadIdx.x & 31;
  const int wave = threadIdx.x >> 5;
  const int lm = lane & 15;
  const int kh = (lane >> 4) << 1;
  const int mo = (lane >> 4) << 3;

  __shared__ float sc[16][264];
  __shared__ float red[8][256];

  const float scale = 0.25f;  // 1/sqrt(16)

  // ---- Phase 1: scores (each wave: 2 key tiles of 16) ----
#pragma unroll
  for (int j = 0; j < 2; ++j) {
    const int k0 = ((wave << 1) + j) << 4;
    v8f acc = {};
    const float* Qr = Q  + (size_t)(base + q0 + lm) * D + hoff + kh;
    const float* Kr = Km + (size_t)(base + k0 + lm) * D + hoff + kh;
#pragma unroll
    for (int kk = 0; kk < 16; kk += 4) {
      v2f a;  a.x  = Qr[kk]; a.y  = Qr[kk + 1];
      v2f bb; bb.x = Kr[kk]; bb.y = Kr[kk + 1];
      acc = WMMA_F32(a, bb, acc);
    }
    const int keyn = k0 + lm;
    const float msk = kept[base + keyn];
#pragma unroll
    for (int r = 0; r < 8; ++r) {
      float v = acc[r] * scale;
      sc[mo + r][keyn] = (msk > 0.5f) ? v : -1e9f;
    }
  }
  __syncthreads();

  // ---- Phase 2: softmax over 256 keys (wave handles rows 2w, 2w+1) ----
#pragma unroll
  for (int j = 0; j < 2; ++j) {
    const int row = (wave << 1) + j;
    float vloc[8];
    float mx = -3.0e38f;
#pragma unroll
    for (int c = 0; c < 8; ++c) {
      vloc[c] = sc[row][lane + (c << 5)];
      mx = fmaxf(mx, vloc[c]);
    }
#pragma unroll
    for (int o = 16; o > 0; o >>= 1) mx = fmaxf(mx, __shfl_xor(mx, o, 32));
    float sum = 0.0f;
#pragma unroll
    for (int c = 0; c < 8; ++c) {
      vloc[c] = expf(vloc[c] - mx);
      sum += vloc[c];
    }
#pragma unroll
    for (int o = 16; o > 0; o >>= 1) sum += __shfl_xor(sum, o, 32);
    const float inv = 1.0f / sum;
#pragma unroll
    for (int c = 0; c < 8; ++c) sc[row][lane + (c << 5)] = vloc[c] * inv;
  }
  __syncthreads();

  // ---- Phase 3: P @ V, keys split across waves (32 keys each) ----
  v8f acc = {};
  const int kb = wave << 5;
#pragma unroll
  for (int kk = 0; kk < 32; kk += 4) {
    const int kr = kb + kk + kh;
    v2f a;  a.x = sc[lm][kr]; a.y = sc[lm][kr + 1];
    const float* Vr = V + (size_t)(base + kr) * D + hoff + lm;
    v2f bb; bb.x = Vr[0]; bb.y = Vr[D];
    acc = WMMA_F32(a, bb, acc);
  }
#pragma unroll
  for (int r = 0; r < 8; ++r) red[wave][(mo + r) * 16 + lm] = acc[r];
  __syncthreads();

  const int t = threadIdx.x;
  float s = 0.0f;
#pragma unroll
  for (int w = 0; w < 8; ++w) s += red[w][t];
  AO[(size_t)(base + q0 + (t >> 4)) * D + hoff + (t & 15)] = s;
}

// ---------------------------------------------------------------------------
// Fused residual LayerNorm (one 256-thread block per token row, D = 512).
// If coef != nullptr: OUT[row] += coef[row] * LN(X+R) (final MoE accumulate),
// else Y[row] = LN(X+R).
// ---------------------------------------------------------------------------
__global__ void ln_kernel(const float* __restrict__ X, const float* __restrict__ R,
                          const float* __restrict__ g, const float* __restrict__ beta,
                          float* __restrict__ Y, const float* __restrict__ coef,
                          float* __restrict__ OUT, int D) {
  const int row = blockIdx.x;
  const int t = threadIdx.x;
  const size_t o0 = (size_t)row * D + t;
  const size_t o1 = o0 + 256;
  const float v0 = X[o0] + R[o0];
  const float v1 = X[o1] + R[o1];
  __shared__ float s1[256], s2[256];
  s1[t] = v0 + v1;
  s2[t] = v0 * v0 + v1 * v1;
  __syncthreads();
  for (int o = 128; o > 0; o >>= 1) {
    if (t < o) { s1[t] += s1[t + o]; s2[t] += s2[t + o]; }
    __syncthreads();
  }
  const float mean = s1[0] / (float)D;
  const float var = s2[0] / (float)D - mean * mean;
  const float inv = rsqrtf(var + 1e-5f);
  const float y0 = (v0 - mean) * inv * g[t] + beta[t];
  const float y1 = (v1 - mean) * inv * g[t + 256] + beta[t + 256];
  if (coef) {
    const float c = coef[row];
    OUT[o0] += c * y0;
    OUT[o1] += c * y1;
  } else {
    Y[o0] = y0;
    Y[o1] = y1;
  }
}

__global__ void addemb(const float* __restrict__ x, const float* __restrict__ emb,
                       float* __restrict__ xe, int D) {
  const int row = blockIdx.x;
  const int t = threadIdx.x;
  const size_t o0 = (size_t)row * D + t;
  xe[o0] = x[o0] + emb[t];
  xe[o0 + 256] = x[o0 + 256] + emb[t + 256];
}

// logits[N,8] = h[N,R] @ Wg[R,8] + bg  (tiny; plain VALU)
__global__ void logits_kernel(const float* __restrict__ h, const float* __restrict__ Wg,
                              const float* __restrict__ bg, float* __restrict__ lgt,
                              int R, int E) {
  const int t = threadIdx.x;
  const int tok = blockIdx.x * 32 + (t >> 3);
  const int e = t & 7;
  float s = bg[e];
  const float* hr = h + (size_t)tok * R;
  for (int r = 0; r < R; ++r) s += hr[r] * Wg[r * E + e];
  lgt[tok * E + e] = s;
}

// ---------------------------------------------------------------------------
// Routing: top-2 + softmax weights, then per-expert capacity keep-first-cap
// via exact sequential-order prefix sum (wave32 ballot + hierarchical scan).
// Single block of 1024 threads (one per token, token order == flat order).
// ---------------------------------------------------------------------------
__global__ void route_kernel(const float* __restrict__ lgt, float* __restrict__ keptf,
                             float* __restrict__ coef, int N, int E) {
  const int t = threadIdx.x;
  const int lane = t & 31;
  const int wid = t >> 5;

  float lg[8];
#pragma unroll
  for (int j = 0; j < 8; ++j) lg[j] = lgt[t * 8 + j];
  int i0 = 0; float v0 = lg[0];
#pragma unroll
  for (int j = 1; j < 8; ++j) if (lg[j] > v0) { v0 = lg[j]; i0 = j; }
  int i1 = -1; float v1 = -3.0e38f;
#pragma unroll
  for (int j = 0; j < 8; ++j) if (j != i0 && lg[j] > v1) { v1 = lg[j]; i1 = j; }
  const float e1 = expf(v1 - v0);           // v0 >= v1
  const float wt0 = 1.0f / (1.0f + e1);
  const float wt1 = e1 / (1.0f + e1);

  __shared__ int wsum[32];
  __shared__ int wpre[32];
  __shared__ int tot;

  for (int e = 0; e < E; ++e) {
    const bool m0 = (i0 == e);
    const bool m1 = (i1 == e);
    const int flag = (m0 || m1) ? 1 : 0;
    const unsigned long long mask = __ballot(flag);
    const int inwave = __popcll(mask & (((unsigned long long)1 << lane) - 1ull));
    if (lane == 0) wsum[wid] = __popcll(mask);
    __syncthreads();
    if (wid == 0) {
      const int v = wsum[lane];
      int incl = v;
#pragma unroll
      for (int o = 1; o < 32; o <<= 1) {
        const int nb = __shfl_up(incl, o, 32);
        if (lane >= o) incl += nb;
      }
      wpre[lane] = incl - v;
      if (lane == 31) tot = incl;
    }
    __syncthreads();
    const int rank = wpre[wid] + inwave;    // exclusive rank in token order
    const int cap = (int)floorf(1.25f * (float)tot / (float)E);
    const bool kept = flag && (rank < cap);
    const float wgt = m0 ? wt0 : (m1 ? wt1 : 0.0f);
    keptf[e * N + t] = kept ? 1.0f : 0.0f;
    coef[e * N + t] = kept ? wgt : 0.0f;
    __syncthreads();
  }
}

__global__ void zero_kernel(float* __restrict__ p, int n) {
  const int i = blockIdx.x * blockDim.x + threadIdx.x;
  if (i < n) p[i] = 0.0f;
}

// ---------------------------------------------------------------------------
extern "C" void kernel_launch(void* const* d_in, const int* in_sizes, int n_in,
                              void* d_out, int out_size, void* d_ws, size_t ws_size,
                              hipStream_t stream) {
  (void)in_sizes; (void)n_in; (void)out_size; (void)ws_size;
  const int Bb = 4, S = 256, D = 512, R = 256, E = 8, H = 32, F = 2048;
  const int N = Bb * S;

  const float* x    = (const float*)d_in[0];
  const float* W_in = (const float*)d_in[1];
  const float* b_in = (const float*)d_in[2];
  // d_in[3] (tau) and d_in[4] (A_m) are unused: h0 = 0 -> h = 0.1*tanh(xp@B_m)
  const float* B_m  = (const float*)d_in[5];
  const float* Wg   = (const float*)d_in[6];
  const float* bg   = (const float*)d_in[7];
  const float* emb  = (const float*)d_in[8];
  const float* Wq   = (const float*)d_in[9];
  const float* bq   = (const float*)d_in[10];
  const float* Wk   = (const float*)d_in[11];
  const float* bk   = (const float*)d_in[12];
  const float* Wv   = (const float*)d_in[13];
  const float* bv   = (const float*)d_in[14];
  const float* Wo   = (const float*)d_in[15];
  const float* bo   = (const float*)d_in[16];
  const float* W1   = (const float*)d_in[17];
  const float* b1   = (const float*)d_in[18];
  const float* W2   = (const float*)d_in[19];
  const float* b2   = (const float*)d_in[20];
  const float* ln1g = (const float*)d_in[21];
  const float* ln1b = (const float*)d_in[22];
  const float* ln2g = (const float*)d_in[23];
  const float* ln2b = (const float*)d_in[24];
  float* out = (float*)d_out;

  float* ws = (float*)d_ws;
  size_t off = 0;
  auto alloc = [&](size_t n) { float* p = ws + off; off += n; return p; };
  float* xp    = alloc((size_t)N * R);
  float* hbuf  = alloc((size_t)N * R);
  float* lgt   = alloc((size_t)N * E);
  float* keptf = alloc((size_t)E * N);
  float* coef  = alloc((size_t)E * N);
  float* xe    = alloc((size_t)N * D);
  float* Qb    = alloc((size_t)N * D);
  float* Kb    = alloc((size_t)N * D);
  float* Vb    = alloc((size_t)N * D);
  float* ao    = alloc((size_t)N * D);
  float* proj  = alloc((size_t)N * D);
  float* yb    = alloc((size_t)N * D);
  float* gb    = alloc((size_t)N * F);
  float* fb    = alloc((size_t)N * D);

  zero_kernel<<<(N * D + 255) / 256, 256, 0, stream>>>(out, N * D);

  auto blocks = [](int M, int Nn) { return (M >> 7) * (Nn >> 6); };

  // ---- Router ----
  gemm128x64<0><<<blocks(N, R), 256, 0, stream>>>(x, W_in, b_in, xp, N, R, D);
  gemm128x64<1><<<blocks(N, R), 256, 0, stream>>>(xp, B_m, nullptr, hbuf, N, R, R);
  logits_kernel<<<N / 32, 256, 0, stream>>>(hbuf, Wg, bg, lgt, R, E);
  route_kernel<<<1, N, 0, stream>>>(lgt, keptf, coef, N, E);

  // ---- Experts ----
  for (int e = 0; e < E; ++e) {
    const size_t eDD = (size_t)e * D * D;
    const size_t eDF = (size_t)e * D * F;
    addemb<<<N, 256, 0, stream>>>(x, emb + (size_t)e * D, xe, D);
    gemm128x64<0><<<blocks(N, D), 256, 0, stream>>>(xe, Wq + eDD, bq + (size_t)e * D, Qb, N, D, D);
    gemm128x64<0><<<blocks(N, D), 256, 0, stream>>>(xe, Wk + eDD, bk + (size_t)e * D, Kb, N, D, D);
    gemm128x64<0><<<blocks(N, D), 256, 0, stream>>>(xe, Wv + eDD, bv + (size_t)e * D, Vb, N, D, D);
    attn16<<<Bb * H * (S >> 4), 256, 0, stream>>>(Qb, Kb, Vb, keptf + (size_t)e * N, ao, S, D, H);
    gemm128x64<0><<<blocks(N, D), 256, 0, stream>>>(ao, Wo + eDD, bo + (size_t)e * D, proj, N, D, D);
    ln_kernel<<<N, 256, 0, stream>>>(xe, proj, ln1g + (size_t)e * D, ln1b + (size_t)e * D,
                                     yb, nullptr, nullptr, D);
    gemm128x64<2><<<blocks(N, F), 256, 0, stream>>>(yb, W1 + eDF, b1 + (size_t)e * F, gb, N, F, D);
    gemm128x64<0><<<blocks(N, D), 256, 0, stream>>>(gb, W2 + eDF, b2 + (size_t)e * D, fb, N, D, F);
    ln_kernel<<<N, 256, 0, stream>>>(yb, fb, ln2g + (size_t)e * D, ln2b + (size_t)e * D,
                                     nullptr, coef + (size_t)e * N, out, D);
  }
}